// LUTLayer_73349451481618
// MI455X (gfx1250) — compile-verified
//
#include <hip/hip_runtime.h>
#include <hip/hip_bf16.h>
#include <float.h>

// MI455X / gfx1250 LUT-layer kernel.
//
// Strategy (see analysis): one workgroup (256 thr = 8 waves, wave32) per (b,s).
//  Phase A: x[b,s,:] (2KB) -> LDS; each thread argmins 2 detectors over 32
//           anchor pairs (int4-vectorized anchor loads), writes the three
//           index/delta outputs, and stashes premultiplied row offsets in LDS.
//  Phase B: L2-bandwidth-bound gather-sum of 512 weight rows (weights = 32MB,
//           L2-resident). Accumulation runs on the matrix pipe via
//           V_WMMA_F32_16X16X4_F32 with an all-ones A matrix:
//           C[m,n] += sum_k B[k,n]  -- full f32 precision, 4 rows x 16 cols
//           per instruction, 4 independent accumulator chains per wave.

typedef __attribute__((ext_vector_type(2))) float v2f;
typedef __attribute__((ext_vector_type(8))) float v8f;

#define B_DIM 8
#define S_DIM 128
#define N_IN  512
#define N_OUT 512
#define D_DIM 512
#define A_DIM 32

__global__ __launch_bounds__(256) void lut_layer_kernel(
    const float* __restrict__ x,            // [B,S,N_IN]
    const float* __restrict__ weights,      // [D*A, N_OUT]
    const int*   __restrict__ anchor_pairs, // [D,A,2]
    float* __restrict__ out_y,              // [B,S,N_OUT]
    int*   __restrict__ out_lookup,         // [B,S,D]
    float* __restrict__ out_mind,           // [B,S,D]
    int*   __restrict__ out_aidx)           // [B,S,D]
{
    __shared__ float xs[N_IN];       // x row for this (b,s)
    __shared__ int   rowoff[D_DIM];  // winning row * N_OUT (element offset)

    const int bs  = blockIdx.x;      // 0 .. B*S-1
    const int tid = threadIdx.x;

    // ---- load x[b,s,:] into LDS (256 threads x float2) ----
    {
        const float2* xrow = (const float2*)(x + (size_t)bs * N_IN);
        ((float2*)xs)[tid] = xrow[tid];
    }
    __syncthreads();

    // ---- Phase A: per-detector argmin over 32 anchors (2 detectors/thread) ----
#pragma unroll
    for (int r = 0; r < 2; ++r) {
        const int d = tid + r * 256;
        const int4* ap = (const int4*)(anchor_pairs + (size_t)d * A_DIM * 2);
        float bestAbs   = FLT_MAX;
        float bestDelta = 0.0f;
        int   bestA     = 0;
#pragma unroll 4
        for (int q = 0; q < A_DIM / 2; ++q) {   // 2 pairs per int4
            int4 p = ap[q];
            float d0 = xs[p.x] - xs[p.y];
            float d1 = xs[p.z] - xs[p.w];
            float a0 = fabsf(d0), a1 = fabsf(d1);
            // strict '<' keeps the FIRST minimum (jnp.argmin semantics)
            if (a0 < bestAbs) { bestAbs = a0; bestDelta = d0; bestA = 2 * q; }
            if (a1 < bestAbs) { bestAbs = a1; bestDelta = d1; bestA = 2 * q + 1; }
        }
        const int    gidx = d * A_DIM + bestA;
        const size_t o    = (size_t)bs * D_DIM + d;
        out_lookup[o] = gidx;
        out_mind[o]   = bestDelta;
        out_aidx[o]   = bestA;
        rowoff[d]     = gidx * N_OUT;
    }
    __syncthreads();

    // ---- Phase B: WMMA ones-trick gather-sum ----
    const int lane  = tid & 31;
    const int wv    = tid >> 5;              // wave 0..7 -> cols [wv*64, wv*64+63]
    const int n     = lane & 15;             // column within 16-wide tile
    const int khalf = (lane >> 4) << 1;      // lanes 0-15 -> K rows {0,1}; 16-31 -> {2,3}
    const int colBase = wv * 64 + n;

    v2f ones; ones.x = 1.0f; ones.y = 1.0f;  // A = 16x4 all-ones
    v8f c0 = {}, c1 = {}, c2 = {}, c3 = {};

#pragma unroll 2
    for (int kb = 0; kb < D_DIM; kb += 4) {
        // two premultiplied row offsets for this lane's K slots (8B-aligned LDS)
        int2 rr = *(const int2*)&rowoff[kb + khalf];
        const float* r0 = weights + rr.x;
        const float* r1 = weights + rr.y;
        v2f b0, b1, b2, b3;
        b0.x = r0[colBase +  0]; b0.y = r1[colBase +  0];
        b1.x = r0[colBase + 16]; b1.y = r1[colBase + 16];
        b2.x = r0[colBase + 32]; b2.y = r1[colBase + 32];
        b3.x = r0[colBase + 48]; b3.y = r1[colBase + 48];
        c0 = __builtin_amdgcn_wmma_f32_16x16x4_f32(false, ones, false, b0, (short)0, c0, false, false);
        c1 = __builtin_amdgcn_wmma_f32_16x16x4_f32(false, ones, false, b1, (short)0, c1, false, false);
        c2 = __builtin_amdgcn_wmma_f32_16x16x4_f32(false, ones, false, b2, (short)0, c2, false, false);
        c3 = __builtin_amdgcn_wmma_f32_16x16x4_f32(false, ones, false, b3, (short)0, c3, false, false);
    }

    // C/D layout: VGPR0 of lanes 0-15 holds row M=0, N = lane. All M rows are
    // identical (ones-A), so lanes 0-15 carry the full column sums.
    if (lane < 16) {
        float* yrow = out_y + (size_t)bs * N_OUT;
        yrow[colBase +  0] = c0[0];
        yrow[colBase + 16] = c1[0];
        yrow[colBase + 32] = c2[0];
        yrow[colBase + 48] = c3[0];
    }
}

extern "C" void kernel_launch(void* const* d_in, const int* in_sizes, int n_in,
                              void* d_out, int out_size, void* d_ws, size_t ws_size,
                              hipStream_t stream) {
    (void)in_sizes; (void)n_in; (void)out_size; (void)d_ws; (void)ws_size;

    const float* x       = (const float*)d_in[0];   // [B,S,N_IN] f32
    const float* weights = (const float*)d_in[1];   // [D*A, N_OUT] f32
    const int*   pairs   = (const int*)d_in[2];     // [D,A,2] i32

    const size_t nBS   = (size_t)B_DIM * S_DIM;     // 1024
    const size_t nOut0 = nBS * N_OUT;               // 524288 (output)
    const size_t nBSD  = nBS * D_DIM;               // 524288 (per index array)

    // d_out = [output f32 | lookup_indices i32 | min_anchor_deltas f32 | a_idx i32]
    float* out_y      = (float*)d_out;
    int*   out_lookup = (int*)((float*)d_out + nOut0);
    float* out_mind   = (float*)d_out + nOut0 + nBSD;
    int*   out_aidx   = (int*)((float*)d_out + nOut0 + 2 * nBSD);

    dim3 grid((unsigned)nBS);   // one workgroup per (b,s)
    dim3 block(256);            // 8 wave32 waves
    lut_layer_kernel<<<grid, block, 0, stream>>>(x, weights, pairs,
                                                 out_y, out_lookup, out_mind, out_aidx);
}